// LIFSpike_29265907155563
// MI455X (gfx1250) — compile-verified
//
#include <hip/hip_runtime.h>
#include <cstdint>
#include <cstddef>

// LIF spike scan: out[e, t] for t in 0..7:
//   u = 0.25*u*(1-o_prev) + x[e,t];  o = (u > 0.2) ? 1 : 0
// Layout: T=8 is innermost/contiguous -> each thread streams its own 32B.

#define THREADS 256
#define GRID    2048
#define T_STEPS 8

typedef __attribute__((ext_vector_type(4))) float f4;
typedef __attribute__((ext_vector_type(4))) int   v4i;

#if defined(__HIP_DEVICE_COMPILE__)

// Builtin signature (from hipcc diagnostic): first param is AS1 `v4i*`,
// second is LDS-side `v4i*`, then imm offset + imm cpol.
typedef __attribute__((address_space(1))) v4i* gv4i_p;
typedef __attribute__((address_space(3))) v4i* lv4i_p;

#if __has_builtin(__builtin_amdgcn_global_load_async_to_lds_b128)
#define ASYNC_B128(g, l, off)                                              \
  __builtin_amdgcn_global_load_async_to_lds_b128(                          \
      (gv4i_p)(uintptr_t)(g),                                              \
      (lv4i_p)(uint32_t)(uintptr_t)(l),                                    \
      off, 0)
#else
#define ASYNC_B128(g, l, off)                                              \
  asm volatile("global_load_async_to_lds_b128 %0, %1, off offset:" #off    \
               :: "v"((uint32_t)(uintptr_t)(l)),                           \
                  "v"((uint64_t)(uintptr_t)(g))                            \
               : "memory")
#endif

#if __has_builtin(__builtin_amdgcn_s_wait_asynccnt)
#define WAIT_ASYNC2() __builtin_amdgcn_s_wait_asynccnt(2)
#else
#define WAIT_ASYNC2() asm volatile("s_wait_asynccnt 2" ::: "memory")
#endif

#else  // host pass: bodies are never codegen'd
#define ASYNC_B128(g, l, off) ((void)0)
#define WAIT_ASYNC2()         ((void)0)
#endif

__global__ __launch_bounds__(THREADS)
void lif_scan_kernel(const float* __restrict__ x, float* __restrict__ out,
                     int nTiles) {
  // per-wave private double buffer: wave w owns smem[b][w*32*8 .. +256)
  __shared__ float smem[2][THREADS * T_STEPS];

  const int tid    = threadIdx.x;
  const int stride = gridDim.x;
  int t = blockIdx.x;
  if (t >= nTiles) return;

  float* l0 = &smem[0][tid * T_STEPS];
  float* l1 = &smem[1][tid * T_STEPS];

  // prologue: stage first tile into buffer 0 (2 x 16B per lane)
  {
    const float* g0 = x + ((size_t)t * THREADS + tid) * T_STEPS;
    ASYNC_B128(g0, l0, 0);
    ASYNC_B128(g0, l0, 16);
  }

  int buf = 0;
  for (; t < nTiles; t += stride) {
    // prefetch next tile into the other buffer; last iteration re-issues a
    // redundant in-range copy so ASYNCcnt bookkeeping stays constant.
    const int nxt = t + stride;
    const int pf  = (nxt < nTiles) ? nxt : t;
    {
      const float* gp = x + ((size_t)pf * THREADS + tid) * T_STEPS;
      float* ln = buf ? l0 : l1;
      ASYNC_B128(gp, ln, 0);
      ASYNC_B128(gp, ln, 16);
    }

    // wait until only the 2 just-issued copies are outstanding ->
    // the current tile's data is resident in LDS.
    WAIT_ASYNC2();
    asm volatile("" ::: "memory");

    const float* l = buf ? l1 : l0;
    f4 a = *(const f4*)(l);
    f4 b = *(const f4*)(l + 4);

    float r[T_STEPS] = {a.x, a.y, a.z, a.w, b.x, b.y, b.z, b.w};
    float u = 0.0f, o = 0.0f;
#pragma unroll
    for (int i = 0; i < T_STEPS; ++i) {
      u    = __builtin_fmaf(0.25f * u, 1.0f - o, r[i]);  // TAU*u*(1-o) + x
      o    = (u > 0.2f) ? 1.0f : 0.0f;                   // spike(u)
      r[i] = o;
    }

    f4 oa = {r[0], r[1], r[2], r[3]};
    f4 ob = {r[4], r[5], r[6], r[7]};
    float* op = out + ((size_t)t * THREADS + tid) * T_STEPS;
    // streaming output, 256MB > 192MB L2 -> non-temporal
    __builtin_nontemporal_store(oa, (f4*)op);
    __builtin_nontemporal_store(ob, (f4*)(op + 4));

    buf ^= 1;
  }
}

// scalar tail safety net (not hit for the reference shape: 4,194,304 % 256 == 0)
__global__ void lif_tail_kernel(const float* __restrict__ x,
                                float* __restrict__ out,
                                int firstElem, int nElems) {
  int e = firstElem + blockIdx.x * blockDim.x + threadIdx.x;
  if (e >= nElems) return;
  const float* g = x + (size_t)e * T_STEPS;
  float* op = out + (size_t)e * T_STEPS;
  float u = 0.0f, o = 0.0f;
#pragma unroll
  for (int i = 0; i < T_STEPS; ++i) {
    u = __builtin_fmaf(0.25f * u, 1.0f - o, g[i]);
    o = (u > 0.2f) ? 1.0f : 0.0f;
    op[i] = o;
  }
}

extern "C" void kernel_launch(void* const* d_in, const int* in_sizes, int n_in,
                              void* d_out, int out_size, void* d_ws, size_t ws_size,
                              hipStream_t stream) {
  const float* x = (const float*)d_in[0];
  float* out     = (float*)d_out;

  const long long nFloats = (long long)in_sizes[0];   // 32*128*32*32*8
  const int nElems = (int)(nFloats / T_STEPS);        // 4,194,304 spatial elems
  const int nTiles = nElems / THREADS;                // 16,384 tiles of 8KB

  if (nTiles > 0) {
    int grid = GRID;
    if (grid > nTiles) grid = nTiles;
    hipLaunchKernelGGL(lif_scan_kernel, dim3(grid), dim3(THREADS), 0, stream,
                       x, out, nTiles);
  }

  const int tailStart = nTiles * THREADS;
  const int tailCount = nElems - tailStart;
  if (tailCount > 0) {
    int tgrid = (tailCount + THREADS - 1) / THREADS;
    hipLaunchKernelGGL(lif_tail_kernel, dim3(tgrid), dim3(THREADS), 0, stream,
                       x, out, tailStart, nElems);
  }
}